// DeformableTransformer_42743514530156
// MI455X (gfx1250) — compile-verified
//
#include <hip/hip_runtime.h>
#include <hip/hip_bf16.h>
#include <math.h>
#include <stdint.h>

// ---------------- problem constants (match reference) ----------------
#define B_   4
#define Q_   900
#define D_   256
#define NH_  8
#define DH_  32
#define L_   4
#define P_   4
#define DFF_ 1024
#define S_   21760   // 128*128 + 64*64 + 32*32 + 16*16

typedef __attribute__((ext_vector_type(2))) float v2f;
typedef __attribute__((ext_vector_type(8))) float v8f;

// ---------------- CDNA5 fp32 WMMA: D(16x16) = A(16x4) * B(4x16) + C ----------------
__device__ __forceinline__ v8f wmma4(v2f a, v2f b, v8f c) {
  // 8 args: (neg_a, A, neg_b, B, c_mod, C, reuse_a, reuse_b)
  return __builtin_amdgcn_wmma_f32_16x16x4_f32(false, a, false, b, (short)0, c,
                                               false, false);
}

// LDS byte offset of a __shared__ object: generic LDS pointers are
// {aperture_hi32 | offset_lo32} (ISA 10.2), so the low 32 bits are the offset.
__device__ __forceinline__ uint32_t lds_off(const void* p) {
  return (uint32_t)(uintptr_t)p;
}

// Async global -> LDS, 16 bytes per lane (GLOBAL_LOAD_ASYNC_TO_LDS_B128,
// tracked by ASYNCcnt). No VGPR data movement; LDS written directly.
__device__ __forceinline__ void async_load_b128(uint32_t lds_addr, const void* gaddr) {
  asm volatile("global_load_async_to_lds_b128 %0, %1, off"
               :: "v"(lds_addr), "v"(gaddr)
               : "memory");
}
__device__ __forceinline__ void wait_asynccnt0() {
  asm volatile("s_wait_asynccnt 0x0" ::: "memory");
}

// =====================================================================
// Generic WMMA GEMM:  out[M][N] = act( X[M][K] @ Wt[N][K]^T + bias[N] )
// Tile: 64(M) x 64(N); K chunks of 32 double-buffered in LDS via
// async-to-LDS loads. 256 threads = 8 waves; wave owns a 16x32 sub-tile.
// =====================================================================
#define TK_  32
#define LDP_ 36   // padded LDS row stride (floats): 16B aligned, conflict-free

__global__ __launch_bounds__(256) void gemm_bias_kernel(
    const float* __restrict__ X, const float* __restrict__ Wt,
    const float* __restrict__ bias, float* __restrict__ out,
    int M, int N, int K, int relu) {
  __shared__ float As[2][64][LDP_];
  __shared__ float Bs[2][64][LDP_];

  const int tid  = threadIdx.x;
  const int lane = tid & 31;
  const int wid  = tid >> 5;
  const int m0 = blockIdx.x * 64;
  const int n0 = blockIdx.y * 64;

  const int mbase = (wid & 3) * 16;
  const int nbase = (wid >> 2) * 32;

  const int l15 = lane & 15;
  const int hi  = lane >> 4;   // 0: lanes 0-15, 1: lanes 16-31
  const int kb  = hi * 2;      // A/B fragment K pair base per ISA layout

  // Per-thread tile-load coordinates: 64 rows x 32 floats = 512 float4,
  // thread handles rows (tid>>3) and (tid>>3)+32 at column (tid&7)*4.
  // Out-of-range rows are CLAMPED (duplicates only feed never-stored outputs).
  const int rowL = tid >> 3;            // 0..31
  const int c4   = (tid & 7) * 4;       // 0..28
  int gmA0 = m0 + rowL;       if (gmA0 >= M) gmA0 = M - 1;
  int gmA1 = m0 + rowL + 32;  if (gmA1 >= M) gmA1 = M - 1;
  int gnB0 = n0 + rowL;       if (gnB0 >= N) gnB0 = N - 1;
  int gnB1 = n0 + rowL + 32;  if (gnB1 >= N) gnB1 = N - 1;

  auto issue = [&](int buf, int k0) {
    async_load_b128(lds_off(&As[buf][rowL][c4]),      &X[(size_t)gmA0 * K + k0 + c4]);
    async_load_b128(lds_off(&As[buf][rowL + 32][c4]), &X[(size_t)gmA1 * K + k0 + c4]);
    async_load_b128(lds_off(&Bs[buf][rowL][c4]),      &Wt[(size_t)gnB0 * K + k0 + c4]);
    async_load_b128(lds_off(&Bs[buf][rowL + 32][c4]), &Wt[(size_t)gnB1 * K + k0 + c4]);
  };

  v8f acc0 = {};
  v8f acc1 = {};

  const int nchunks = K / TK_;   // K is a multiple of 32 for all layers here
  issue(0, 0);
  wait_asynccnt0();
  __syncthreads();

  for (int c = 0; c < nchunks; ++c) {
    const int cur = c & 1;
    if (c + 1 < nchunks) issue(cur ^ 1, (c + 1) * TK_);   // prefetch next chunk

#pragma unroll
    for (int kk = 0; kk < TK_; kk += 4) {
      v2f a, b0, b1;
      a.x  = As[cur][mbase + l15][kk + kb];
      a.y  = As[cur][mbase + l15][kk + kb + 1];
      b0.x = Bs[cur][nbase + l15][kk + kb];
      b0.y = Bs[cur][nbase + l15][kk + kb + 1];
      b1.x = Bs[cur][nbase + 16 + l15][kk + kb];
      b1.y = Bs[cur][nbase + 16 + l15][kk + kb + 1];
      acc0 = wmma4(a, b0, acc0);
      acc1 = wmma4(a, b1, acc1);
    }

    wait_asynccnt0();   // our async writes into the other buffer have landed
    __syncthreads();    // everyone done reading cur + writing nxt
  }

  // epilogue: C/D layout -> lane holds col (lane&15), VGPR i holds row i / i+8
#pragma unroll
  for (int i = 0; i < 8; ++i) {
    int m = m0 + mbase + hi * 8 + i;
    if (m < M) {
      int n  = n0 + nbase + l15;
      float v0 = acc0[i] + bias[n];
      float v1 = acc1[i] + bias[n + 16];
      if (relu) { v0 = fmaxf(v0, 0.f); v1 = fmaxf(v1, 0.f); }
      out[(size_t)m * N + n]      = v0;
      out[(size_t)m * N + n + 16] = v1;
    }
  }
}

// =====================================================================
// Fused MHA self-attention (flash style), one wave per (b, head, 16-q tile)
// scores = (Q/sqrt(DH)) @ K^T via WMMA, online softmax, O += P @ V via WMMA.
// Tail rows/keys handled by clamped loads + column masking.
// =====================================================================
__global__ __launch_bounds__(32) void mha_kernel(const float* __restrict__ qkv,
                                                 float* __restrict__ attn) {
  __shared__ float Qs[16][LDP_];
  __shared__ float Ks[16][LDP_];
  __shared__ float Vs[16][LDP_];
  __shared__ float Ps[16][20];

  const int lane = threadIdx.x;
  const int QT = (Q_ + 15) / 16;      // 57 tiles
  const int qt = blockIdx.x % QT;
  const int h  = (blockIdx.x / QT) % NH_;
  const int b  = blockIdx.x / (QT * NH_);
  const int q0 = qt * 16;

  const int l15 = lane & 15;
  const int hi  = lane >> 4;
  const int kb  = hi * 2;
  const float scale = 0.17677669529663687f;   // 1/sqrt(32)

  // ---- load Q tile (scaled); clamp rows, stores are guarded later ----
#pragma unroll
  for (int j = 0; j < 4; ++j) {
    int f4 = lane + 32 * j;
    int row = f4 >> 3;
    int c4 = (f4 & 7) * 4;
    int gq = q0 + row; if (gq >= Q_) gq = Q_ - 1;
    float4 v = *reinterpret_cast<const float4*>(
        &qkv[(size_t)(b * Q_ + gq) * (3 * D_) + h * DH_ + c4]);
    v.x *= scale; v.y *= scale; v.z *= scale; v.w *= scale;
    *reinterpret_cast<float4*>(&Qs[row][c4]) = v;
  }

  float mrow[8], lrow[8];
  v8f o0 = {};
  v8f o1 = {};
#pragma unroll
  for (int i = 0; i < 8; ++i) { mrow[i] = -3.0e38f; lrow[i] = 0.f; }

  for (int kt = 0; kt < QT; ++kt) {
    int k0 = kt * 16;
    // ---- load K, V tiles; clamped rows are masked out of the softmax ----
#pragma unroll
    for (int j = 0; j < 4; ++j) {
      int f4 = lane + 32 * j;
      int row = f4 >> 3;
      int c4 = (f4 & 7) * 4;
      int gk = k0 + row; if (gk >= Q_) gk = Q_ - 1;
      const float* base = &qkv[(size_t)(b * Q_ + gk) * (3 * D_)];
      float4 vk = *reinterpret_cast<const float4*>(&base[D_ + h * DH_ + c4]);
      float4 vv = *reinterpret_cast<const float4*>(&base[2 * D_ + h * DH_ + c4]);
      *reinterpret_cast<float4*>(&Ks[row][c4]) = vk;
      *reinterpret_cast<float4*>(&Vs[row][c4]) = vv;
    }
    __syncthreads();

    // ---- S = Q * K^T (contract over DH=32, 8 chained WMMAs) ----
    v8f s = {};
#pragma unroll
    for (int kk = 0; kk < 32; kk += 4) {
      v2f a, bb;
      a.x  = Qs[l15][kk + kb];
      a.y  = Qs[l15][kk + kb + 1];
      bb.x = Ks[l15][kk + kb];
      bb.y = Ks[l15][kk + kb + 1];
      s = wmma4(a, bb, s);
    }
    // mask out-of-range key columns (lane's column index = l15)
    bool kvalid = (k0 + l15) < Q_;
#pragma unroll
    for (int i = 0; i < 8; ++i)
      if (!kvalid) s[i] = -3.0e38f;

    // ---- online softmax (row reductions across 16-lane halves) ----
#pragma unroll
    for (int i = 0; i < 8; ++i) {
      float mi = s[i];
#pragma unroll
      for (int o = 8; o >= 1; o >>= 1) mi = fmaxf(mi, __shfl_xor(mi, o, 32));
      float mn    = fmaxf(mrow[i], mi);
      float alpha = __expf(mrow[i] - mn);
      float p     = __expf(s[i] - mn);
      float rs    = p;
#pragma unroll
      for (int o = 8; o >= 1; o >>= 1) rs += __shfl_xor(rs, o, 32);
      lrow[i] = lrow[i] * alpha + rs;
      mrow[i] = mn;
      o0[i] *= alpha;
      o1[i] *= alpha;
      Ps[hi * 8 + i][l15] = p;   // D-layout -> LDS for A-layout regather
    }
    __syncthreads();

    // ---- O += P (16x16) * V (16x32), contract over 16 keys ----
#pragma unroll
    for (int kk = 0; kk < 16; kk += 4) {
      v2f a, b0, b1;
      a.x  = Ps[l15][kk + kb];
      a.y  = Ps[l15][kk + kb + 1];
      b0.x = Vs[kk + kb][l15];
      b0.y = Vs[kk + kb + 1][l15];
      b1.x = Vs[kk + kb][16 + l15];
      b1.y = Vs[kk + kb + 1][16 + l15];
      o0 = wmma4(a, b0, o0);
      o1 = wmma4(a, b1, o1);
    }
    __syncthreads();
  }

  // ---- normalize and store (head-major output layout) ----
#pragma unroll
  for (int i = 0; i < 8; ++i) {
    int m = q0 + hi * 8 + i;
    if (m < Q_) {
      float inv = 1.0f / lrow[i];
      float* dst = &attn[(size_t)(b * Q_ + m) * D_ + h * DH_];
      dst[l15]      = o0[i] * inv;
      dst[16 + l15] = o1[i] * inv;
    }
  }
}

// =====================================================================
// Residual + LayerNorm over D=256: one wave per row, 8 elems per lane.
// =====================================================================
__global__ __launch_bounds__(256) void ln_res_kernel(
    const float* __restrict__ x, const float* __restrict__ r,
    const float* __restrict__ g, const float* __restrict__ be,
    float* __restrict__ out, int M) {
  int row  = blockIdx.x * 8 + (threadIdx.x >> 5);
  int lane = threadIdx.x & 31;
  if (row >= M) return;
  const float* xr = x + (size_t)row * D_;
  const float* rr = r + (size_t)row * D_;
  int c0 = lane * 4, c1 = 128 + lane * 4;

  float4 a0 = *reinterpret_cast<const float4*>(&xr[c0]);
  float4 a1 = *reinterpret_cast<const float4*>(&xr[c1]);
  float4 b0 = *reinterpret_cast<const float4*>(&rr[c0]);
  float4 b1 = *reinterpret_cast<const float4*>(&rr[c1]);
  float v[8] = {a0.x + b0.x, a0.y + b0.y, a0.z + b0.z, a0.w + b0.w,
                a1.x + b1.x, a1.y + b1.y, a1.z + b1.z, a1.w + b1.w};

  float s = 0.f, s2 = 0.f;
#pragma unroll
  for (int i = 0; i < 8; ++i) { s += v[i]; s2 += v[i] * v[i]; }
#pragma unroll
  for (int o = 16; o >= 1; o >>= 1) {
    s  += __shfl_xor(s, o, 32);
    s2 += __shfl_xor(s2, o, 32);
  }
  float mean = s * (1.0f / D_);
  float var  = s2 * (1.0f / D_) - mean * mean;
  float rstd = rsqrtf(var + 1e-5f);

  float4 g0 = *reinterpret_cast<const float4*>(&g[c0]);
  float4 g1 = *reinterpret_cast<const float4*>(&g[c1]);
  float4 e0 = *reinterpret_cast<const float4*>(&be[c0]);
  float4 e1 = *reinterpret_cast<const float4*>(&be[c1]);
  float4 o0, o1;
  o0.x = (v[0] - mean) * rstd * g0.x + e0.x;
  o0.y = (v[1] - mean) * rstd * g0.y + e0.y;
  o0.z = (v[2] - mean) * rstd * g0.z + e0.z;
  o0.w = (v[3] - mean) * rstd * g0.w + e0.w;
  o1.x = (v[4] - mean) * rstd * g1.x + e1.x;
  o1.y = (v[5] - mean) * rstd * g1.y + e1.y;
  o1.z = (v[6] - mean) * rstd * g1.z + e1.z;
  o1.w = (v[7] - mean) * rstd * g1.w + e1.w;
  *reinterpret_cast<float4*>(&out[(size_t)row * D_ + c0]) = o0;
  *reinterpret_cast<float4*>(&out[(size_t)row * D_ + c1]) = o1;
}

// =====================================================================
// Multi-scale deformable sampling: one wave per (b, q, head); lanes = DH.
// =====================================================================
__global__ __launch_bounds__(256) void msdeform_kernel(
    const float* __restrict__ ref, const float* __restrict__ value,
    const float* __restrict__ off, const float* __restrict__ awl,
    const unsigned char* __restrict__ mask, float* __restrict__ samp) {
  int wid  = blockIdx.x * 8 + (threadIdx.x >> 5);
  int lane = threadIdx.x & 31;
  if (wid >= B_ * Q_ * NH_) return;
  int h = wid % NH_;
  int q = (wid / NH_) % Q_;
  int b = wid / (NH_ * Q_);
  size_t bq = (size_t)b * Q_ + q;

  // softmax over the 16 (level,point) logits: lanes 0..15 hold one each
  float logit = (lane < 16) ? awl[bq * (NH_ * 16) + h * 16 + lane] : -3.0e38f;
  float mx = logit;
#pragma unroll
  for (int o = 8; o >= 1; o >>= 1) mx = fmaxf(mx, __shfl_xor(mx, o, 32));
  float e = __expf(logit - mx);
  float sm = e;
#pragma unroll
  for (int o = 8; o >= 1; o >>= 1) sm += __shfl_xor(sm, o, 32);
  float w = e / sm;

  constexpr int LW[4]  = {128, 64, 32, 16};
  constexpr int LH[4]  = {128, 64, 32, 16};
  constexpr int LST[4] = {0, 16384, 20480, 21504};

  float acc = 0.f;
#pragma unroll
  for (int j = 0; j < 16; ++j) {
    const int l = j >> 2, p = j & 3;
    const float Wl = (float)LW[l], Hl = (float)LH[l];
    float wj = __shfl(w, j, 32);
    float ox = off[bq * 256 + h * 32 + l * 8 + p * 2 + 0];
    float oy = off[bq * 256 + h * 32 + l * 8 + p * 2 + 1];
    float rx = ref[(bq * 4 + l) * 2 + 0];
    float ry = ref[(bq * 4 + l) * 2 + 1];
    float x = (rx + ox / Wl) * Wl - 0.5f;
    float y = (ry + oy / Hl) * Hl - 0.5f;
    float xf = floorf(x), yf = floorf(y);
    float dx = x - xf, dy = y - yf;
    int x0 = (int)xf, y0 = (int)yf;
#pragma unroll
    for (int c = 0; c < 4; ++c) {
      int oxi = c & 1, oyi = c >> 1;
      int xi = x0 + oxi, yi = y0 + oyi;
      float wc = (oxi ? dx : 1.f - dx) * (oyi ? dy : 1.f - dy);
      bool valid = (xi >= 0) && (xi < LW[l]) && (yi >= 0) && (yi < LH[l]);
      int xic = min(max(xi, 0), LW[l] - 1);
      int yic = min(max(yi, 0), LH[l] - 1);
      int sidx = LST[l] + yic * LW[l] + xic;
      if (mask[(size_t)b * S_ + sidx]) valid = false;
      float gv = value[((size_t)b * S_ + sidx) * D_ + h * DH_ + lane];
      acc += wj * (valid ? wc : 0.f) * gv;
    }
  }
  samp[bq * D_ + h * DH_ + lane] = acc;
}

// =====================================================================
// Host launcher
// =====================================================================
extern "C" void kernel_launch(void* const* d_in, const int* in_sizes, int n_in,
                              void* d_out, int out_size, void* d_ws, size_t ws_size,
                              hipStream_t stream) {
  (void)in_sizes; (void)n_in; (void)out_size; (void)ws_size;

  const float* tgt  = (const float*)d_in[0];
  const float* ref  = (const float*)d_in[1];
  const float* src  = (const float*)d_in[2];
  // d_in[3] spatial shapes, d_in[4] level starts: compile-time constants here
  const unsigned char* mask = (const unsigned char*)d_in[5];
  const float* sa_in_w  = (const float*)d_in[6];
  const float* sa_in_b  = (const float*)d_in[7];
  const float* sa_out_w = (const float*)d_in[8];
  const float* sa_out_b = (const float*)d_in[9];
  const float* off_w    = (const float*)d_in[10];
  const float* off_b    = (const float*)d_in[11];
  const float* aw_w     = (const float*)d_in[12];
  const float* aw_b     = (const float*)d_in[13];
  const float* val_w    = (const float*)d_in[14];
  const float* val_b    = (const float*)d_in[15];
  const float* co_w     = (const float*)d_in[16];
  const float* co_b     = (const float*)d_in[17];
  const float* ln1_g    = (const float*)d_in[18];
  const float* ln1_b    = (const float*)d_in[19];
  const float* ln2_g    = (const float*)d_in[20];
  const float* ln2_b    = (const float*)d_in[21];
  const float* ln3_g    = (const float*)d_in[22];
  const float* ln3_b    = (const float*)d_in[23];
  const float* ffn_w1   = (const float*)d_in[24];
  const float* ffn_b1   = (const float*)d_in[25];
  const float* ffn_w2   = (const float*)d_in[26];
  const float* ffn_b2   = (const float*)d_in[27];

  float* ws = (float*)d_ws;
  const size_t NBQ = (size_t)B_ * Q_;   // 3600 rows

  // persistent region
  float* x1   = ws;                     // (B,Q,D)
  float* offb = x1 + NBQ * D_;          // (B,Q,256)
  float* awb  = offb + NBQ * D_;        // (B,Q,128)
  float* samp = awb + NBQ * 128;        // (B,Q,D)
  float* co   = samp + NBQ * D_;        // (B,Q,D)
  float* x2   = co + NBQ * D_;          // (B,Q,D)
  // union region (phases don't overlap in time)
  float* un   = x2 + NBQ * D_;
  float* qkv  = un;                     // phase 1: (B,Q,768)
  float* attn = un + NBQ * 3 * D_;      //          (B,Q,256)
  float* mha  = attn + NBQ * D_;        //          (B,Q,256)
  float* valu = un;                     // phase 2: (B,S,256)  85 MB
  float* ffh  = un;                     // phase 3: (B,Q,1024)
  float* ffo  = un + NBQ * DFF_;        //          (B,Q,256)

  const int M = (int)NBQ;
  dim3 blk(256);
  const int QT = (Q_ + 15) / 16;

  // 1. qkv projection
  gemm_bias_kernel<<<dim3((M + 63) / 64, (3 * D_) / 64), blk, 0, stream>>>(
      tgt, sa_in_w, sa_in_b, qkv, M, 3 * D_, D_, 0);
  // 2. fused self-attention
  mha_kernel<<<dim3(B_ * NH_ * QT), dim3(32), 0, stream>>>(qkv, attn);
  // 3. output projection
  gemm_bias_kernel<<<dim3((M + 63) / 64, D_ / 64), blk, 0, stream>>>(
      attn, sa_out_w, sa_out_b, mha, M, D_, D_, 0);
  // 4. x1 = LN(tgt + mha)
  ln_res_kernel<<<dim3((M + 7) / 8), blk, 0, stream>>>(mha, tgt, ln2_g, ln2_b, x1, M);
  // 5. value projection over all source tokens
  gemm_bias_kernel<<<dim3((B_ * S_ + 63) / 64, D_ / 64), blk, 0, stream>>>(
      src, val_w, val_b, valu, B_ * S_, D_, D_, 0);
  // 6. sampling offsets
  gemm_bias_kernel<<<dim3((M + 63) / 64, D_ / 64), blk, 0, stream>>>(
      x1, off_w, off_b, offb, M, D_, D_, 0);
  // 7. attention weight logits
  gemm_bias_kernel<<<dim3((M + 63) / 64, 128 / 64), blk, 0, stream>>>(
      x1, aw_w, aw_b, awb, M, 128, D_, 0);
  // 8. deformable bilinear sampling
  msdeform_kernel<<<dim3((B_ * Q_ * NH_ + 7) / 8), blk, 0, stream>>>(
      ref, valu, offb, awb, mask, samp);
  // 9. cross-attn output projection
  gemm_bias_kernel<<<dim3((M + 63) / 64, D_ / 64), blk, 0, stream>>>(
      samp, co_w, co_b, co, M, D_, D_, 0);
  // 10. x2 = LN(x1 + co)
  ln_res_kernel<<<dim3((M + 7) / 8), blk, 0, stream>>>(co, x1, ln1_g, ln1_b, x2, M);
  // 11. FFN up + ReLU
  gemm_bias_kernel<<<dim3((M + 63) / 64, DFF_ / 64), blk, 0, stream>>>(
      x2, ffn_w1, ffn_b1, ffh, M, DFF_, D_, 1);
  // 12. FFN down
  gemm_bias_kernel<<<dim3((M + 63) / 64, D_ / 64), blk, 0, stream>>>(
      ffh, ffn_w2, ffn_b2, ffo, M, D_, DFF_, 0);
  // 13. out = LN(x2 + ff)
  ln_res_kernel<<<dim3((M + 7) / 8), blk, 0, stream>>>(
      ffo, x2, ln3_g, ln3_b, (float*)d_out, M);
}